// Hausdorff_Distance_35416300323043
// MI455X (gfx1250) — compile-verified
//
#include <hip/hip_runtime.h>
#include <hip/hip_bf16.h>

#define B_  8
#define C_  4
#define H_  96
#define W_  96
#define HW_ (H_ * W_)            // 9216
#define NT_ (HW_ / 16)           // 576 column tiles
#define NPAIR_ (B_ * C_)         // 32
#define BIG_ 1e30f

typedef __attribute__((ext_vector_type(2))) float v2f;
typedef __attribute__((ext_vector_type(4))) float v4f;
typedef __attribute__((ext_vector_type(8))) float v8f;

// Order-preserving float <-> uint mapping so we can use integer atomicMax.
__device__ __forceinline__ unsigned f2ord(float f) {
    unsigned u = __float_as_uint(f);
    return (u & 0x80000000u) ? ~u : (u | 0x80000000u);
}
__device__ __forceinline__ float ord2f(unsigned u) {
    u = (u & 0x80000000u) ? (u ^ 0x80000000u) : ~u;
    return __uint_as_float(u);
}

// Kernel 1: per-pixel class byte (pred argmax low nibble, label argmax high
// nibble) + init the 32 ordered-uint max accumulators.
__global__ void hd_prep(const float* __restrict__ pred,
                        const float* __restrict__ lbl,
                        unsigned char* __restrict__ cls,
                        unsigned* __restrict__ acc) {
    int idx = blockIdx.x * blockDim.x + threadIdx.x;
    if (idx < NPAIR_) acc[idx] = f2ord(-BIG_);
    if (idx >= B_ * HW_) return;
    int b = idx / HW_, p = idx % HW_;
    const float* pp = pred + (size_t)b * C_ * HW_ + p;
    float best = pp[0];
    int bi = 0;
#pragma unroll
    for (int c = 1; c < C_; ++c) {
        float v = pp[(size_t)c * HW_];
        if (v > best) { best = v; bi = c; }   // first-max wins, like argmax
    }
    const float* lp = lbl + (size_t)b * C_ * HW_ + p;
    int li = 0;
#pragma unroll
    for (int c = 0; c < C_; ++c)
        if (lp[(size_t)c * HW_] > 0.5f) li = c;
    cls[idx] = (unsigned char)(bi | (li << 4));
}

// Kernel 2: build the interleaved per-(pair, q) B-fragment table. Column masks
// are FOLDED INTO the r2 channel (masked-out -> r2+BIG, and 1e30 + O(1e4) ==
// 1e30 in fp32, so the WMMA output is exactly BIG for masked columns).
// Layout (16 B per (q, half) entry, 32 B per q):
//   entry(q, half=0) = (1, r2+addA, 1, r2+addB)   -> K{0,1} of both directions
//   entry(q, half=1) = (x, y,       x, y      )   -> K{2,3} of both directions
// One global_load_b128 per lane per tile yields BOTH direction fragments.
__global__ void hd_tabs(const unsigned char* __restrict__ cls,
                        float* __restrict__ itab) {
    int idx = blockIdx.x * blockDim.x + threadIdx.x;   // pair*HW + q
    if (idx >= NPAIR_ * HW_) return;
    int pair = idx / HW_;
    int q = idx - pair * HW_;
    int b = pair >> 2, cc = pair & 3;
    float qy = (float)(q / W_);
    float qx = (float)(q % W_);
    float r2 = qx * qx + qy * qy;
    unsigned char mc = cls[b * HW_ + q];
    float addA = ((mc & 0xF) == cc) ? 0.0f : BIG_;  // pred-set mask
    float addB = ((mc >> 4)  == cc) ? 0.0f : BIG_;  // label-set mask
    v4f e0 = {1.0f, r2 + addA, 1.0f, r2 + addB};
    v4f e1 = {qx, qy, qx, qy};
    v4f* dst = (v4f*)itab + (size_t)idx * 2;
    dst[0] = e0;
    dst[1] = e1;
}

// Kernel 3: one wave per (pair, 16-row block). Per 16x16 tile: one b128 load
// (both direction fragments), two V_WMMA_F32_16X16X4_F32, 16 running mins
// (compiler fuses across the x2 unroll into v_min3_num_f32).
__global__ void __launch_bounds__(32) hd_tiles(const unsigned char* __restrict__ cls,
                                               const float* __restrict__ itab,
                                               unsigned* __restrict__ acc) {
    const int lane = threadIdx.x;      // full wave32, EXEC all ones
    const int half = lane >> 4;
    const int l    = lane & 15;
    const int rowBlk = blockIdx.x;     // 0..575
    const int pair   = blockIdx.y;     // 0..31 == b*C + c
    const int cc = pair & 3;

    // Uniform (SGPR) base + 32-bit per-lane index -> saddr+voffset loads.
    const v4f* tpair = (const v4f*)itab + (size_t)pair * HW_ * 2;
    const int li = l * 2 + half;       // per-lane entry index; +32 per tile

    // A fragment: lane l holds row m=l; vgpr0 = K(0|2), vgpr1 = K(1|3).
    const int P0 = rowBlk * 16;
    const int p  = P0 + l;
    const float py = (float)(p / W_);
    const float px = (float)(p % W_);
    v2f afrag;
    afrag.x = half ? (-2.0f * px) : (px * px + py * py);
    afrag.y = half ? (-2.0f * py) : 1.0f;

    float rminA[8], rminB[8];
#pragma unroll
    for (int v = 0; v < 8; ++v) { rminA[v] = BIG_; rminB[v] = BIG_; }

#pragma unroll 2
    for (int qb = 0; qb < NT_; ++qb) {
        const v4f w = tpair[li + qb * 32];       // one b128: both fragments
        v2f bA; bA.x = w.x; bA.y = w.y;          // pred-masked  K pair
        v2f bB; bB.x = w.z; bB.y = w.w;          // label-masked K pair

        v8f dA = {};
        dA = __builtin_amdgcn_wmma_f32_16x16x4_f32(
                 false, afrag, false, bA, (short)0, dA, false, false);
        v8f dB = {};
        dB = __builtin_amdgcn_wmma_f32_16x16x4_f32(
                 false, afrag, false, bB, (short)0, dB, false, false);
#pragma unroll
        for (int v = 0; v < 8; ++v) {
            rminA[v] = fminf(rminA[v], dA[v]);  // min_{q in pred set} d2
            rminB[v] = fminf(rminB[v], dB[v]);  // min_{q in label set} d2
        }
    }

    // Row-min across the 16 lanes sharing each row (C/D layout: M = v + 8*half).
#pragma unroll
    for (int v = 0; v < 8; ++v) {
#pragma unroll
        for (int off = 1; off < 16; off <<= 1) {
            rminA[v] = fminf(rminA[v], __shfl_xor(rminA[v], off, 32));
            rminB[v] = fminf(rminB[v], __shfl_xor(rminB[v], off, 32));
        }
    }

    if (l == 0) {                      // lanes 0 and 16 each own 8 rows
        const unsigned char* mcls = cls + (size_t)(pair >> 2) * HW_;
        float localMax = -BIG_;
#pragma unroll
        for (int v = 0; v < 8; ++v) {
            const int pr = P0 + v + half * 8;
            const unsigned char mc = mcls[pr];
            const float cB = ((mc & 0xF) == cc) ? rminB[v] : -BIG_; // p in A -> minB
            const float cA = ((mc >> 4)  == cc) ? rminA[v] : -BIG_; // p in B -> minA
            localMax = fmaxf(localMax, fmaxf(cA, cB));
        }
        atomicMax(&acc[pair], f2ord(localMax));
    }
}

// Kernel 4: decode accumulators, sqrt, means, write [B, C+2].
__global__ void hd_final(const unsigned* __restrict__ acc, float* __restrict__ out) {
    const int b = threadIdx.x;
    if (b >= B_) return;
    float hd[C_];
    float s = 0.0f, sf = 0.0f;
#pragma unroll
    for (int c = 0; c < C_; ++c) {
        const float d2 = ord2f(acc[b * C_ + c]);
        float h = (d2 < 0.0f) ? -BIG_ : sqrtf(d2);
        if (d2 >= BIG_) h = BIG_;      // empty-target degenerate case
        hd[c] = h;
        s += h;
        if (c > 0) sf += h;
    }
#pragma unroll
    for (int c = 0; c < C_; ++c) out[b * (C_ + 2) + c] = hd[c];
    out[b * (C_ + 2) + C_]     = s  * (1.0f / C_);
    out[b * (C_ + 2) + C_ + 1] = sf * (1.0f / (C_ - 1));
}

extern "C" void kernel_launch(void* const* d_in, const int* in_sizes, int n_in,
                              void* d_out, int out_size, void* d_ws, size_t ws_size,
                              hipStream_t stream) {
    const float* pred = (const float*)d_in[0];   // [B, C, H, W] fp32
    const float* lbl  = (const float*)d_in[1];   // [B, C, H, W] fp32 one-hot
    float* out = (float*)d_out;                  // [B, C+2] fp32

    // Workspace layout (256-aligned):
    //   cls : B*HW bytes        =    73,728
    //   acc : 32 uints (+pad)   =       256
    //   itab: 32*HW*32 bytes    = 9,437,184   (total ~9.1 MB)
    unsigned char* cls = (unsigned char*)d_ws;
    unsigned* acc = (unsigned*)((char*)d_ws + (size_t)(B_ * HW_));
    float* itab = (float*)((char*)d_ws + (size_t)(B_ * HW_) + 256);

    const int n = B_ * HW_;
    hd_prep<<<(n + 255) / 256, 256, 0, stream>>>(pred, lbl, cls, acc);

    const int nt = NPAIR_ * HW_;
    hd_tabs<<<(nt + 255) / 256, 256, 0, stream>>>(cls, itab);

    dim3 grid(NT_, NPAIR_);                      // 576 row blocks x 32 pairs
    hd_tiles<<<grid, 32, 0, stream>>>(cls, itab, acc);

    hd_final<<<1, 32, 0, stream>>>(acc, out);
}